// GCN_72653666779673
// MI455X (gfx1250) — compile-verified
//
#include <hip/hip_runtime.h>
#include <hip/hip_bf16.h>
#include <math.h>

typedef __attribute__((ext_vector_type(2))) float v2f;
typedef __attribute__((ext_vector_type(8))) float v8f;

#define GCN_NODES 100000
#define GCN_EDGES 3200000
#define GCN_INF   256

// ---------------------------------------------------------------------------
// Degree counting: out_deg[src] += 1, in_deg[dst] += 1 (fp32, like segment_sum of ones)
// ---------------------------------------------------------------------------
__global__ __launch_bounds__(256)
void gcn_degree_count(const int* __restrict__ src, const int* __restrict__ dst,
                      float* __restrict__ outdeg, float* __restrict__ indeg, int E) {
    for (int i = blockIdx.x * blockDim.x + threadIdx.x; i < E;
         i += gridDim.x * blockDim.x) {
        atomicAdd(&outdeg[src[i]], 1.0f);
        atomicAdd(&indeg[dst[i]], 1.0f);
    }
}

// norm = clip(deg,1)^-0.5, in place on both arrays
__global__ __launch_bounds__(256)
void gcn_degree_norm(float* __restrict__ a, float* __restrict__ b, int n) {
    int i = blockIdx.x * blockDim.x + threadIdx.x;
    if (i < n) {
        a[i] = 1.0f / sqrtf(fmaxf(a[i], 1.0f));
        b[i] = 1.0f / sqrtf(fmaxf(b[i], 1.0f));
    }
}

// ---------------------------------------------------------------------------
// WMMA fp32 GEMM:  C[M,N] = (A[M,K] * scale[m]) @ W[K,N]
// One wave per 16-row stripe; NT = number of 16-col tiles (covers N<=NT*16).
// Uses V_WMMA_F32_16X16X4_F32 (wave32, fp32 matrix pipe).
// ---------------------------------------------------------------------------
template<int NT>
__global__ __launch_bounds__(256)
void gcn_gemm_wmma(const float* __restrict__ A, const float* __restrict__ W,
                   const float* __restrict__ scale, float* __restrict__ C,
                   int M, int N, int K) {
    const int lane  = threadIdx.x & 31;
    const int wv    = threadIdx.x >> 5;
    const int mtile = blockIdx.x * (blockDim.x >> 5) + wv;
    const int row0  = mtile * 16;
    if (row0 >= M) return;                       // wave-uniform: EXEC stays all-1s

    const int ml = lane & 15;                    // row within tile (A) / col within tile (B,C)
    const int kk = (lane >> 4) << 1;             // K sub-offset: 0 or 2

    int arow = row0 + ml; if (arow >= M) arow = M - 1;   // clamp (not stored anyway)
    const float s = scale[arow];
    const float* __restrict__ aptr = A + (size_t)arow * K + kk;

    int bcol[NT];
#pragma unroll
    for (int t = 0; t < NT; ++t) {
        int c = t * 16 + ml;
        bcol[t] = (c < N) ? c : (N - 1);         // clamp OOB column loads
    }

    v8f acc[NT] = {};

    for (int k0 = 0; k0 < K; k0 += 4) {
        v2f a;
        a.x = aptr[k0]     * s;                  // A[arow][k0+kk]
        a.y = aptr[k0 + 1] * s;                  // A[arow][k0+kk+1]
        const float* __restrict__ wrow = W + (size_t)(k0 + kk) * N;
#pragma unroll
        for (int t = 0; t < NT; ++t) {
            v2f b;
            b.x = wrow[bcol[t]];                 // W[k0+kk][col]
            b.y = wrow[bcol[t] + N];             // W[k0+kk+1][col]
            acc[t] = __builtin_amdgcn_wmma_f32_16x16x4_f32(
                false, a, false, b, (short)0, acc[t], false, false);
        }
    }

    // D layout: VGPR r holds row (r + 8*(lane>>4)), column (lane&15)
    const int rbase = row0 + ((lane >> 4) << 3);
#pragma unroll
    for (int t = 0; t < NT; ++t) {
        int c = t * 16 + ml;
        if (c < N) {
#pragma unroll
            for (int r = 0; r < 8; ++r) {
                int rr = rbase + r;
                if (rr < M) C[(size_t)rr * N + c] = acc[t][r];
            }
        }
    }
}

// ---------------------------------------------------------------------------
// SpMM scatter: agg[dst[e]] += h[src[e]]  (edge-parallel, float4 gather + atomics)
// ---------------------------------------------------------------------------
__global__ __launch_bounds__(256)
void gcn_spmm_scatter(const int* __restrict__ src, const int* __restrict__ dst,
                      const float* __restrict__ h, float* __restrict__ agg,
                      int E, int F) {
    const int F4 = F >> 2;
    const long long total  = (long long)E * F4;
    const long long stride = (long long)gridDim.x * blockDim.x;
    for (long long t = (long long)blockIdx.x * blockDim.x + threadIdx.x; t < total;
         t += stride) {
        int e = (int)(t / F4);
        int c = (int)(t - (long long)e * F4) << 2;
        int sn = src[e], dn = dst[e];
        const float4 v = *reinterpret_cast<const float4*>(h + (size_t)sn * F + c);
        float* o = agg + (size_t)dn * F + c;
        atomicAdd(o + 0, v.x);
        atomicAdd(o + 1, v.y);
        atomicAdd(o + 2, v.z);
        atomicAdd(o + 3, v.w);
    }
}

// h[n][f] = relu(agg[n][f] * norm_d[n] + b[f])
__global__ __launch_bounds__(256)
void gcn_bias_relu_norm(const float* __restrict__ agg, const float* __restrict__ normd,
                        const float* __restrict__ b, float* __restrict__ h,
                        int M, int F) {
    size_t total = (size_t)M * F;
    for (size_t i = (size_t)blockIdx.x * blockDim.x + threadIdx.x; i < total;
         i += (size_t)gridDim.x * blockDim.x) {
        size_t n = i / F;
        int    f = (int)(i - n * F);
        h[i] = fmaxf(agg[i] * normd[n] + b[f], 0.0f);
    }
}

// ---------------------------------------------------------------------------
// Attention pooling
// ---------------------------------------------------------------------------
__global__ __launch_bounds__(320)
void gcn_colsum40(const float* __restrict__ h, float* __restrict__ colsum, int M) {
    __shared__ float part[320];
    const int c = threadIdx.x % 40;
    const int w = threadIdx.x / 40;   // 0..7
    float acc = 0.0f;
    for (int n = blockIdx.x * 8 + w; n < M; n += gridDim.x * 8)
        acc += h[(size_t)n * 40 + c];
    part[threadIdx.x] = acc;
    __syncthreads();
    if (threadIdx.x < 40) {
        float s = 0.0f;
        for (int w2 = 0; w2 < 8; ++w2) s += part[w2 * 40 + c];
        atomicAdd(&colsum[c], s);
    }
}

// tvec[j] = tanh( (colsum/N) @ Wa )[j]
__global__ void gcn_context40(const float* __restrict__ colsum, const float* __restrict__ Wa,
                              float* __restrict__ tvec, float invM) {
    int j = threadIdx.x;
    if (j < 40) {
        float s = 0.0f;
        for (int i = 0; i < 40; ++i) s += (colsum[i] * invM) * Wa[i * 40 + j];
        tvec[j] = tanhf(s);
    }
}

// rep[j] += sum_n h[n][j] * sigmoid(h[n]·tvec)
__global__ __launch_bounds__(256)
void gcn_attn_rep(const float* __restrict__ h, const float* __restrict__ tvec,
                  float* __restrict__ rep, int M) {
    __shared__ float st[40];
    __shared__ float srep[40];
    if (threadIdx.x < 40) { st[threadIdx.x] = tvec[threadIdx.x]; srep[threadIdx.x] = 0.0f; }
    __syncthreads();
    for (int n = blockIdx.x * blockDim.x + threadIdx.x; n < M;
         n += gridDim.x * blockDim.x) {
        const float* row = h + (size_t)n * 40;
        float dot = 0.0f;
#pragma unroll
        for (int j = 0; j < 40; j += 4) {
            float4 v = *reinterpret_cast<const float4*>(row + j);
            dot += v.x * st[j] + v.y * st[j + 1] + v.z * st[j + 2] + v.w * st[j + 3];
        }
        float sc = 1.0f / (1.0f + expf(-dot));
#pragma unroll
        for (int j = 0; j < 40; j += 4) {
            float4 v = *reinterpret_cast<const float4*>(row + j);
            atomicAdd(&srep[j],     v.x * sc);   // LDS ds_add_f32
            atomicAdd(&srep[j + 1], v.y * sc);
            atomicAdd(&srep[j + 2], v.z * sc);
            atomicAdd(&srep[j + 3], v.w * sc);
        }
    }
    __syncthreads();
    if (threadIdx.x < 40) atomicAdd(&rep[threadIdx.x], srep[threadIdx.x]);
}

// Tiny MLP head: 40 -> 30 -> 10 -> 1 (single wave)
__global__ void gcn_mlp_head(const float* __restrict__ rep,
                             const float* __restrict__ Wm1, const float* __restrict__ bm1,
                             const float* __restrict__ Wm2, const float* __restrict__ bm2,
                             const float* __restrict__ Wm3, const float* __restrict__ bm3,
                             float* __restrict__ out) {
    __shared__ float g1[30];
    __shared__ float g2[10];
    const int t = threadIdx.x;
    if (t < 30) {
        float s = bm1[t];
        for (int j = 0; j < 40; ++j) s += rep[j] * Wm1[j * 30 + t];
        g1[t] = fmaxf(s, 0.0f);
    }
    __syncthreads();
    if (t < 10) {
        float s = bm2[t];
        for (int j = 0; j < 30; ++j) s += g1[j] * Wm2[j * 10 + t];
        g2[t] = fmaxf(s, 0.0f);
    }
    __syncthreads();
    if (t == 0) {
        float s = bm3[0];
        for (int j = 0; j < 10; ++j) s += g2[j] * Wm3[j];
        out[0] = s;
    }
}

// ---------------------------------------------------------------------------
extern "C" void kernel_launch(void* const* d_in, const int* in_sizes, int n_in,
                              void* d_out, int out_size, void* d_ws, size_t ws_size,
                              hipStream_t stream) {
    (void)in_sizes; (void)n_in; (void)out_size; (void)ws_size;

    const float* in_feat = (const float*)d_in[0];
    const int*   src     = (const int*)d_in[1];
    const int*   dst     = (const int*)d_in[2];
    const float* W1      = (const float*)d_in[3];
    const float* b1      = (const float*)d_in[4];
    const float* W2      = (const float*)d_in[5];
    const float* b2      = (const float*)d_in[6];
    const float* Wa      = (const float*)d_in[7];
    const float* Wm1     = (const float*)d_in[8];
    const float* bm1     = (const float*)d_in[9];
    const float* Wm2     = (const float*)d_in[10];
    const float* bm2     = (const float*)d_in[11];
    const float* Wm3     = (const float*)d_in[12];
    const float* bm3     = (const float*)d_in[13];
    float* out = (float*)d_out;

    const int M = GCN_NODES, E = GCN_EDGES;

    // Workspace layout (floats): norm_s[M] | norm_d[M] | h1[80M] | agg[80M] | h2[40M] | colsum[40] tvec[40] rep[40]
    float* ws     = (float*)d_ws;
    float* norm_s = ws;
    float* norm_d = ws + (size_t)M;
    float* h1     = ws + 2  * (size_t)M;
    float* agg    = ws + 82 * (size_t)M;
    float* h2     = ws + 162 * (size_t)M;
    float* colsum = ws + 202 * (size_t)M;
    float* tvec   = colsum + 40;
    float* rep    = tvec + 40;

    // Zero atomic accumulation buffers (graph-capturable memsets)
    hipMemsetAsync(norm_s, 0, 2 * (size_t)M * sizeof(float), stream);
    hipMemsetAsync(colsum, 0, 120 * sizeof(float), stream);

    gcn_degree_count<<<2048, 256, 0, stream>>>(src, dst, norm_s, norm_d, E);
    gcn_degree_norm<<<(M + 255) / 256, 256, 0, stream>>>(norm_s, norm_d, M);

    const int mtiles = (M + 15) / 16;             // 6250
    const int gblocks = (mtiles + 7) / 8;         // 8 waves / block

    // Layer 1: h1 = relu( Dd^-1/2 * Adj * (Ds^-1/2 * x) @ W1 + b1 )
    gcn_gemm_wmma<5><<<gblocks, 256, 0, stream>>>(in_feat, W1, norm_s, h1, M, 80, GCN_INF);
    hipMemsetAsync(agg, 0, (size_t)M * 80 * sizeof(float), stream);
    gcn_spmm_scatter<<<20000, 256, 0, stream>>>(src, dst, h1, agg, E, 80);
    gcn_bias_relu_norm<<<8192, 256, 0, stream>>>(agg, norm_d, b1, h1, M, 80);

    // Layer 2: h2 = relu( Dd^-1/2 * Adj * (Ds^-1/2 * h1) @ W2 + b2 )
    gcn_gemm_wmma<3><<<gblocks, 256, 0, stream>>>(h1, W2, norm_s, h2, M, 40, 80);
    hipMemsetAsync(agg, 0, (size_t)M * 40 * sizeof(float), stream);
    gcn_spmm_scatter<<<20000, 256, 0, stream>>>(src, dst, h2, agg, E, 40);
    gcn_bias_relu_norm<<<8192, 256, 0, stream>>>(agg, norm_d, b2, h2, M, 40);

    // Attention pooling + MLP head
    gcn_colsum40<<<256, 320, 0, stream>>>(h2, colsum, M);
    gcn_context40<<<1, 64, 0, stream>>>(colsum, Wa, tvec, 1.0f / (float)M);
    gcn_attn_rep<<<400, 256, 0, stream>>>(h2, tvec, rep, M);
    gcn_mlp_head<<<1, 32, 0, stream>>>(rep, Wm1, bm1, Wm2, bm2, Wm3, bm3, out);
}